// part_loss_30382598652051
// MI455X (gfx1250) — compile-verified
//
#include <hip/hip_runtime.h>
#include <math.h>

typedef __attribute__((ext_vector_type(2))) float v2f;
typedef __attribute__((ext_vector_type(8))) float v8f;

#define NB 4   // batches
#define NS 4   // segments

// ---------------------------------------------------------------------------
// K0: initialize workspace.
// pts   : B*S*N float4 sentinel points (0,0,0,1e30)  -> d = sqrt(1e30+|a|^2) ~ 1e15
// dmin  : B*S*N uint, FLT_MAX bit pattern (all values are non-negative floats,
//         so uint ordering == float ordering and atomicMin on bits is exact)
// cnt / d1sum / chamfer : per-(b,s) accumulators
// ---------------------------------------------------------------------------
__global__ void k0_init(float4* __restrict__ pts, unsigned* __restrict__ dmin,
                        int* __restrict__ cnt, float* __restrict__ d1sum,
                        float* __restrict__ chamfer, int total) {
    int i = blockIdx.x * blockDim.x + threadIdx.x;
    if (i < total) {
        pts[i]  = make_float4(0.0f, 0.0f, 0.0f, 1.0e30f);
        dmin[i] = 0x7F7FFFFFu;   // FLT_MAX
    }
    if (i < NB * NS) {
        cnt[i] = 0;
        d1sum[i] = 0.0f;
        chamfer[i] = 0.0f;
    }
}

// ---------------------------------------------------------------------------
// K1: partition GT points by label into per-(b,s) contiguous float4 regions,
// precomputing |g|^2 into .w. Order within a segment is irrelevant (min/sum).
// ---------------------------------------------------------------------------
__global__ void k1_scatter(const float* __restrict__ gt, float4* __restrict__ pts,
                           int* __restrict__ cnt, int N) {
    int i = blockIdx.x * blockDim.x + threadIdx.x;
    if (i >= NB * N) return;
    int b = i / N;
    int n = i - b * N;
    const float* g = gt + ((size_t)b * N + n) * 4;
    float x = g[0], y = g[1], z = g[2];
    int s = (int)g[3];
    s = s < 0 ? 0 : (s > NS - 1 ? NS - 1 : s);
    int bs = b * NS + s;
    int pos = atomicAdd(&cnt[bs], 1);
    pts[(size_t)bs * N + pos] = make_float4(x, y, z, x * x + y * y + z * z);
}

// ---------------------------------------------------------------------------
// K2: WMMA core. Grid = B*S*(M/128) blocks, 256 threads = 8 waves.
// Wave w owns 16 pred rows [m0, m0+16). A-fragment (16x4 f32, float2/lane):
//   lanes 0-15  : (K0,K1) = (-2x, -2y) of row M=lane
//   lanes 16-31 : (K2,K3) = (-2z, |a|^2) of row M=lane-16
// B-fragment (4x16 f32):
//   lanes 0-15  : (K0,K1) = (gx, gy) of col N=lane
//   lanes 16-31 : (K2,K3) = (gz, 1.0) of col N=lane-16
// D tile element = |a|^2 - 2 a.g ; add per-column |g|^2, clamp, sqrt.
// The n-loop body is fully uniform straight-line code: lane selection is
// branchless (v_cndmask) and the column-min combine across the two row-halves
// is delegated to the global atomicMin itself (lane L and L+16 hit the same
// address), so there is no EXEC manipulation anywhere near the WMMA.
// ---------------------------------------------------------------------------
__global__ __launch_bounds__(256)
void k2_wmma(const float* __restrict__ ret, const float4* __restrict__ pts,
             const int* __restrict__ cnt, unsigned* __restrict__ dmin,
             float* __restrict__ d1sum, int M, int N, int MT) {
    int mt = blockIdx.x % MT;
    int bs = blockIdx.x / MT;            // b*NS + s
    int lane = threadIdx.x & 31;
    int l16  = lane & 15;
    bool hi  = lane >= 16;
    int m0 = mt * 128 + (threadIdx.x >> 5) * 16;

    // Load this wave's A fragment once (loop-invariant over n-tiles).
    const float* pr = ret + ((size_t)bs * M + (m0 + l16)) * 3;
    float ax = pr[0], ay = pr[1], az = pr[2];
    float a2 = ax * ax + ay * ay + az * az;
    v2f afrag;
    afrag[0] = hi ? (-2.0f * az) : (-2.0f * ax);
    afrag[1] = hi ? a2           : (-2.0f * ay);

    const float4* seg = pts + (size_t)bs * N;
    unsigned* dmb = dmin + (size_t)bs * N;
    int cn = cnt[bs];
    int ntiles = (cn + 15) >> 4;         // block-uniform -> EXEC all-1 at WMMA

    float rowmin[8];
#pragma unroll
    for (int r = 0; r < 8; ++r) rowmin[r] = 1.0e10f;   // reference BIG

    for (int t = 0; t < ntiles; ++t) {
        int n0 = t * 16;
        float4 p = seg[n0 + l16];        // one global_load_b128, no branches
        v2f bfrag;
        bfrag[0] = hi ? p.z  : p.x;
        bfrag[1] = hi ? 1.0f : p.y;
        float g2 = p.w;

        v8f c = {};
        c = __builtin_amdgcn_wmma_f32_16x16x4_f32(
                false, afrag, false, bfrag, (short)0, c, false, false);

        float cm = 3.0e38f;
#pragma unroll
        for (int r = 0; r < 8; ++r) {
            // raw v_sqrt_f32: inputs >= 1e-12, no denormal/overflow fixups needed
            float dv = __builtin_amdgcn_sqrtf(fmaxf(c[r] + g2, 1e-12f));
            rowmin[r] = fminf(rowmin[r], dv);
            cm = fminf(cm, dv);
        }
        // Partial column min (8 rows of this half-wave). Lane L and L+16 write
        // the same column; the atomic merges the two halves and all waves.
        atomicMin(&dmb[n0 + l16], __float_as_uint(cm));
    }

    // Row mins: reduce across the 16 lanes of each half (xor stays in-half).
#pragma unroll
    for (int r = 0; r < 8; ++r) {
        float v = rowmin[r];
        v = fminf(v, __shfl_xor(v, 1, 32));
        v = fminf(v, __shfl_xor(v, 2, 32));
        v = fminf(v, __shfl_xor(v, 4, 32));
        v = fminf(v, __shfl_xor(v, 8, 32));
        rowmin[r] = v;
    }
    if (l16 == 0) {                       // lanes 0 and 16 both active
        float s8 = 0.0f;
#pragma unroll
        for (int r = 0; r < 8; ++r) s8 += rowmin[r];   // 8 rows each half
        s8 += __shfl_xor(s8, 16, 32);
        if (lane == 0) atomicAdd(&d1sum[bs], s8);      // sum of 16 row-mins
    }
}

// ---------------------------------------------------------------------------
// K3: per-(b,s) block: sum column mins over n < cnt, form chamfer term.
// ---------------------------------------------------------------------------
__global__ __launch_bounds__(256)
void k3_seg(const unsigned* __restrict__ dmin, const int* __restrict__ cnt,
            const float* __restrict__ d1sum, float* __restrict__ chamfer,
            int M, int N) {
    int bs = blockIdx.x;
    int cn = cnt[bs];
    const unsigned* dmb = dmin + (size_t)bs * N;
    float sum = 0.0f;
    for (int n = threadIdx.x; n < cn; n += blockDim.x)
        sum += __uint_as_float(dmb[n]);
    __shared__ float sh[256];
    sh[threadIdx.x] = sum;
    __syncthreads();
    for (int off = 128; off > 0; off >>= 1) {
        if ((int)threadIdx.x < off) sh[threadIdx.x] += sh[threadIdx.x + off];
        __syncthreads();
    }
    if (threadIdx.x == 0) {
        float ch = 0.0f;
        if (cn > 0)
            ch = 0.5f * (d1sum[bs] / (float)M + sh[0] / (float)cn);
        chamfer[bs] = ch;
    }
}

// ---------------------------------------------------------------------------
// K4: scalar finalize (16 values).
// ---------------------------------------------------------------------------
__global__ void k4_final(const float* __restrict__ chamfer,
                         const int* __restrict__ cnt, float* __restrict__ out) {
    if (blockIdx.x == 0 && threadIdx.x == 0) {
        float acc = 0.0f;
        for (int b = 0; b < NB; ++b) {
            float cs = 0.0f;
            int present = 0;
            for (int s = 0; s < NS; ++s) {
                cs += chamfer[b * NS + s];
                present += (cnt[b * NS + s] > 0) ? 1 : 0;
            }
            acc += cs / (float)(present > 0 ? present : 1);
        }
        out[0] = acc / (float)NB;
    }
}

extern "C" void kernel_launch(void* const* d_in, const int* in_sizes, int n_in,
                              void* d_out, int out_size, void* d_ws, size_t ws_size,
                              hipStream_t stream) {
    const float* ret = (const float*)d_in[0];         // (B, S*M*3)
    const float* gt  = (const float*)d_in[1];         // (B, N, 4)
    // d_in[2] = seg_num_all (device scalar); S fixed at 4 per problem setup.

    const int M = in_sizes[0] / (NB * NS * 3);        // 1024
    const int N = in_sizes[1] / (NB * 4);             // 8192
    const int MT = M / 128;                           // 8 m-tiles

    // Workspace layout
    char* ws = (char*)d_ws;
    float4*   pts   = (float4*)ws;                    // B*S*N float4
    size_t off = (size_t)NB * NS * N * sizeof(float4);
    unsigned* dmin  = (unsigned*)(ws + off);          // B*S*N uint
    off += (size_t)NB * NS * N * sizeof(unsigned);
    int*      cnt   = (int*)(ws + off);               // 16 ints
    off += 256;
    float*    d1sum = (float*)(ws + off);             // 16 floats
    off += 256;
    float*    chamf = (float*)(ws + off);             // 16 floats

    const int total = NB * NS * N;

    k0_init<<<(total + 255) / 256, 256, 0, stream>>>(pts, dmin, cnt, d1sum, chamf, total);
    k1_scatter<<<(NB * N + 255) / 256, 256, 0, stream>>>(gt, pts, cnt, N);
    k2_wmma<<<NB * NS * MT, 256, 0, stream>>>(ret, pts, cnt, dmin, d1sum, M, N, MT);
    k3_seg<<<NB * NS, 256, 0, stream>>>(dmin, cnt, d1sum, chamf, M, N);
    k4_final<<<1, 32, 0, stream>>>(chamf, cnt, (float*)d_out);
}